// GraphAttentionLayer_68487548502667
// MI455X (gfx1250) — compile-verified
//
#include <hip/hip_runtime.h>
#include <hip/hip_bf16.h>
#include <math.h>
#include <stdint.h>

typedef float v2f __attribute__((ext_vector_type(2)));
typedef float v8f __attribute__((ext_vector_type(8)));
typedef unsigned int u32x4 __attribute__((ext_vector_type(4)));
typedef int i32x8 __attribute__((ext_vector_type(8)));
typedef int i32x4 __attribute__((ext_vector_type(4)));

#define GAT_ALPHA 0.1f
#define GAT_NEG_INF -9000000000000000.0f

static constexpr int GN    = 4096;  // nodes
static constexpr int GFIN  = 2048;
static constexpr int GFOUT = 2048;

// ---------------------------------------------------------------------------
// Tensor Data Mover descriptor helpers (CDNA5 ISA ch.8, D# groups).
// Group0: [1:0]count=1 | [63:32]lds_addr | [120:64]global_addr | [127:126]type=2
// Group1: data_size=2(4B), 2-D tensor dims/strides, optional LDS padding.
// ---------------------------------------------------------------------------
static __device__ __forceinline__ i32x8 tdm_g1_2d(unsigned td0, unsigned td1,
                                                  unsigned tile0, unsigned tile1,
                                                  unsigned long long stride0_elems,
                                                  unsigned pad_en, unsigned pad_interval,
                                                  unsigned pad_amount) {
  i32x8 g;
  g[0] = (int)((2u << 16) | (pad_en << 20) | (pad_interval << 22) | (pad_amount << 25));
  g[1] = (int)((td0 & 0xFFFFu) << 16);                                   // tensor_dim0[15:0]
  g[2] = (int)(((td0 >> 16) & 0xFFFFu) | ((td1 & 0xFFFFu) << 16));       // td0[31:16], td1[15:0]
  g[3] = (int)(((td1 >> 16) & 0xFFFFu) | ((tile0 & 0xFFFFu) << 16));     // td1[31:16], tile_dim0
  g[4] = (int)(tile1 & 0xFFFFu);                                         // tile_dim1, tile_dim2=0
  g[5] = (int)(stride0_elems & 0xFFFFFFFFull);                           // tensor_dim0_stride lo
  g[6] = (int)((stride0_elems >> 32) & 0xFFFFull);                       // stride hi16
  g[7] = 0;
  return g;
}

static __device__ __forceinline__ void tdm_issue(unsigned lds_byte_off,
                                                 unsigned long long gaddr,
                                                 i32x8 g1) {
  u32x4 g0;
  g0.x = 1u;                                        // count=1, user mode
  g0.y = lds_byte_off;                              // lds_addr
  g0.z = (unsigned)gaddr;                           // global_addr[31:0]
  g0.w = (unsigned)(gaddr >> 32) | (2u << 30);      // global_addr[56:32] | type=2
  i32x4 z4 = {0, 0, 0, 0};
  i32x8 z8 = {0, 0, 0, 0, 0, 0, 0, 0};
  // 6-arg toolchain variant: (g0, g1, g2, g3, g4, cpol)
  __builtin_amdgcn_tensor_load_to_lds(g0, g1, z4, z4, z8, 0);
}

// ---------------------------------------------------------------------------
// Kernel 1: h = x @ W   (f32 WMMA 16x16x4, TDM-staged, double buffered)
// 256 thr / 8 waves. Block tile 128(M) x 128(N); wave tile 32 x 64
// (wave = wm*2+wn, wm in 0..3 -> 32-row strip, wn in 0..1 -> 64-col strip).
// K chunk = 16. A tile 128x16 staged with row stride 18 (TDM pad: conflict-free
// + 8B aligned float2 A-fragment reads). B tile 16x128 row-major.
// Fragment maps (f32 WMMA): A vgpr r: M=lo, K=r+2*hi -> float2 @ col kk+2*hi.
//                           B vgpr r: N=lo, K=r+2*hi. C vgpr r: M=r+8*hi, N=lo.
// ---------------------------------------------------------------------------
__global__ __launch_bounds__(256) void gat_gemm_xw(const float* __restrict__ x,
                                                   const float* __restrict__ W,
                                                   float* __restrict__ h) {
  __shared__ float sA[2][128 * 18];
  __shared__ float sB[2][16 * 128];

  const int lane = threadIdx.x & 31;
  const int wave = threadIdx.x >> 5;
  const int lo = lane & 15, hi = lane >> 4;
  const int wm = wave >> 1, wn = wave & 1;
  const int blockM = blockIdx.y * 128;
  const int blockN = blockIdx.x * 128;

  const i32x8 g1A = tdm_g1_2d(GFIN, GN, 16, 128, GFIN, 1u, 3u, 1u);  // pad 2 dw / 16 dw
  const i32x8 g1B = tdm_g1_2d(GFOUT, GFIN, 128, 16, GFOUT, 0u, 0u, 0u);

  const unsigned ldsA[2] = {(unsigned)(uintptr_t)&sA[0][0], (unsigned)(uintptr_t)&sA[1][0]};
  const unsigned ldsB[2] = {(unsigned)(uintptr_t)&sB[0][0], (unsigned)(uintptr_t)&sB[1][0]};
  const unsigned long long baseA = (unsigned long long)(uintptr_t)(x + (size_t)blockM * GFIN);
  const unsigned long long baseB = (unsigned long long)(uintptr_t)(W + (size_t)blockN);

  constexpr int NC = GFIN / 16;                     // 128 K-chunks
  if (wave == 0) {
    tdm_issue(ldsA[0], baseA, g1A);
    tdm_issue(ldsB[0], baseB, g1B);
  }

  v8f c[2][4] = {};
  int buf = 0;
  for (int ch = 0; ch < NC; ++ch) {
    if (ch + 1 < NC) {
      if (wave == 0) {
        tdm_issue(ldsA[buf ^ 1], baseA + (unsigned long long)(ch + 1) * 64ull, g1A);
        tdm_issue(ldsB[buf ^ 1], baseB + (unsigned long long)(ch + 1) * (16ull * GFOUT * 4ull), g1B);
      }
      __builtin_amdgcn_s_wait_tensorcnt((short)2);  // current chunk landed (in-order TDM)
    } else {
      __builtin_amdgcn_s_wait_tensorcnt((short)0);
    }
    __syncthreads();

    const float* __restrict__ pa = &sA[buf][0];
    const float* __restrict__ pb = &sB[buf][0];
#pragma unroll
    for (int kk = 0; kk < 16; kk += 4) {
      v2f a[2], b[4];
#pragma unroll
      for (int s = 0; s < 2; ++s) {
        const int r = 32 * wm + 16 * s + lo;
        const float2 av = *reinterpret_cast<const float2*>(pa + r * 18 + kk + 2 * hi);
        a[s].x = av.x; a[s].y = av.y;                           // ds_load_b64
      }
#pragma unroll
      for (int t = 0; t < 4; ++t) {
        const int cc = 64 * wn + 16 * t + lo;
        b[t].x = pb[(kk + 2 * hi) * 128 + cc];
        b[t].y = pb[(kk + 1 + 2 * hi) * 128 + cc];
      }
#pragma unroll
      for (int s = 0; s < 2; ++s)
#pragma unroll
        for (int t = 0; t < 4; ++t)
          c[s][t] = __builtin_amdgcn_wmma_f32_16x16x4_f32(false, a[s], false, b[t],
                                                          (short)0, c[s][t], false, false);
    }
    __syncthreads();
    buf ^= 1;
  }

#pragma unroll
  for (int s = 0; s < 2; ++s) {
    float* hb = h + (size_t)(blockM + 32 * wm + 16 * s + 8 * hi) * GFOUT + blockN + 64 * wn;
#pragma unroll
    for (int t = 0; t < 4; ++t)
#pragma unroll
      for (int r = 0; r < 8; ++r)
        hb[(size_t)r * GFOUT + 16 * t + lo] = c[s][t][r];
  }
}

// ---------------------------------------------------------------------------
// Kernel 2: s1[i] = h[i,:]@a1 ; s2[i] = h[i,:]@a2  (one block per row)
// ---------------------------------------------------------------------------
__global__ __launch_bounds__(256) void gat_scores(const float* __restrict__ h,
                                                  const float* __restrict__ a,
                                                  float* __restrict__ s1,
                                                  float* __restrict__ s2) {
  const int i = blockIdx.x;
  const float* hr = h + (size_t)i * GFOUT;
  float p1 = 0.f, p2 = 0.f;
  for (int k = threadIdx.x; k < GFOUT; k += 256) {
    const float hv = hr[k];
    p1 += hv * a[k];
    p2 += hv * a[GFOUT + k];
  }
  __shared__ float r1[256], r2[256];
  r1[threadIdx.x] = p1; r2[threadIdx.x] = p2;
  __syncthreads();
  for (int s = 128; s > 0; s >>= 1) {
    if (threadIdx.x < s) {
      r1[threadIdx.x] += r1[threadIdx.x + s];
      r2[threadIdx.x] += r2[threadIdx.x + s];
    }
    __syncthreads();
  }
  if (threadIdx.x == 0) { s1[i] = r1[0]; s2[i] = r2[0]; }
}

// ---------------------------------------------------------------------------
// Kernel 3: per-row online-softmax stats over masked logits.
// Masked value is finite (-9e15) -> isolated rows get uniform softmax, matching
// the reference semantics exactly.
// ---------------------------------------------------------------------------
__global__ __launch_bounds__(256) void gat_softmax_stats(const int* __restrict__ adj,
                                                         const float* __restrict__ s1,
                                                         const float* __restrict__ s2,
                                                         float* __restrict__ mrow,
                                                         float* __restrict__ zrow) {
  const int i = blockIdx.x;
  const float s1i = s1[i];
  const int* ar = adj + (size_t)i * GN;
  float m = -INFINITY, z = 0.f;
  for (int j = threadIdx.x; j < GN; j += 256) {
    __builtin_prefetch(ar + j + 2048, 0, 0);  // global_prefetch_b8
    float v = s1i + s2[j];
    float e = v > 0.f ? v : GAT_ALPHA * v;
    e = (ar[j] > 0) ? e : GAT_NEG_INF;
    if (e > m) { z = z * __expf(m - e) + 1.f; m = e; }
    else       { z += __expf(e - m); }
  }
  __shared__ float sm[256], sz[256];
  sm[threadIdx.x] = m; sz[threadIdx.x] = z;
  __syncthreads();
  for (int s = 128; s > 0; s >>= 1) {
    if (threadIdx.x < s) {
      const float ma = sm[threadIdx.x], mb = sm[threadIdx.x + s];
      const float za = sz[threadIdx.x], zb = sz[threadIdx.x + s];
      const float mm = fmaxf(ma, mb);
      sm[threadIdx.x] = mm;
      sz[threadIdx.x] = za * __expf(ma - mm) + zb * __expf(mb - mm);
    }
    __syncthreads();
  }
  if (threadIdx.x == 0) { mrow[i] = sm[0]; zrow[i] = sz[0]; }
}

// ---------------------------------------------------------------------------
// Kernel 4: out = elu( softmax(e) @ h ). Attention probs generated in registers
// as the WMMA A fragment (v_exp co-executes with WMMA); h rows (B operand)
// TDM-staged into LDS, shared by all 8 waves (8 M-strips), double buffered.
// Block tile 128(M) x 64(N); wave tile 16 x 64. J chunk = 16.
// ---------------------------------------------------------------------------
__global__ __launch_bounds__(256) void gat_aggregate(const int* __restrict__ adj,
                                                     const float* __restrict__ h,
                                                     const float* __restrict__ s1,
                                                     const float* __restrict__ s2,
                                                     const float* __restrict__ mrow,
                                                     const float* __restrict__ zrow,
                                                     float* __restrict__ out) {
  __shared__ float sB[2][16 * 64];

  const int lane = threadIdx.x & 31;
  const int wave = threadIdx.x >> 5;
  const int lo = lane & 15, hi = lane >> 4;
  const int m0 = blockIdx.y * 128 + wave * 16;
  const int blockN = blockIdx.x * 64;
  const int row = m0 + lo;

  const float s1i  = s1[row];
  const float mi   = mrow[row];
  const float invZ = 1.f / zrow[row];
  const int* ar = adj + (size_t)row * GN;

  const i32x8 g1B = tdm_g1_2d(GFOUT, GN, 64, 16, GFOUT, 0u, 0u, 0u);
  const unsigned ldsB[2] = {(unsigned)(uintptr_t)&sB[0][0], (unsigned)(uintptr_t)&sB[1][0]};
  const unsigned long long baseB = (unsigned long long)(uintptr_t)(h + (size_t)blockN);

  constexpr int NC = GN / 16;                       // 256 J-chunks
  if (wave == 0) tdm_issue(ldsB[0], baseB, g1B);

  v8f c[4] = {};
  int buf = 0;
  for (int ch = 0; ch < NC; ++ch) {
    if (ch + 1 < NC) {
      if (wave == 0)
        tdm_issue(ldsB[buf ^ 1], baseB + (unsigned long long)(ch + 1) * (16ull * GFOUT * 4ull), g1B);
      __builtin_amdgcn_s_wait_tensorcnt((short)1);
    } else {
      __builtin_amdgcn_s_wait_tensorcnt((short)0);
    }
    __syncthreads();

    const int j0 = ch * 16;
    __builtin_prefetch(ar + j0 + 1024, 0, 0);       // stream adj ahead
    const float* __restrict__ pb = &sB[buf][0];
#pragma unroll
    for (int kk = 0; kk < 16; kk += 4) {
      const int jA = j0 + kk + 2 * hi;              // K = jA, jA+1 for A vgpr 0,1
      const int2   av = *reinterpret_cast<const int2*>(ar + jA);
      const float2 sv = *reinterpret_cast<const float2*>(s2 + jA);
      float e0 = s1i + sv.x; e0 = e0 > 0.f ? e0 : GAT_ALPHA * e0;
      float e1 = s1i + sv.y; e1 = e1 > 0.f ? e1 : GAT_ALPHA * e1;
      e0 = (av.x > 0) ? e0 : GAT_NEG_INF;
      e1 = (av.y > 0) ? e1 : GAT_NEG_INF;
      v2f a;
      a.x = __expf(e0 - mi) * invZ;
      a.y = __expf(e1 - mi) * invZ;
      v2f b[4];
#pragma unroll
      for (int t = 0; t < 4; ++t) {
        const int cc = 16 * t + lo;
        b[t].x = pb[(kk + 2 * hi) * 64 + cc];
        b[t].y = pb[(kk + 1 + 2 * hi) * 64 + cc];
      }
#pragma unroll
      for (int t = 0; t < 4; ++t)
        c[t] = __builtin_amdgcn_wmma_f32_16x16x4_f32(false, a, false, b[t],
                                                     (short)0, c[t], false, false);
    }
    __syncthreads();
    buf ^= 1;
  }

  float* ob = out + (size_t)(m0 + 8 * hi) * GFOUT + blockN;
#pragma unroll
  for (int t = 0; t < 4; ++t)
#pragma unroll
    for (int r = 0; r < 8; ++r) {
      const float v = c[t][r];
      ob[(size_t)r * GFOUT + 16 * t + lo] = v > 0.f ? v : __expf(v) - 1.f;  // ELU
    }
}

// ---------------------------------------------------------------------------
extern "C" void kernel_launch(void* const* d_in, const int* in_sizes, int n_in,
                              void* d_out, int out_size, void* d_ws, size_t ws_size,
                              hipStream_t stream) {
  const float* x   = (const float*)d_in[0];
  const int*   adj = (const int*)d_in[1];
  const float* W   = (const float*)d_in[2];
  const float* a   = (const float*)d_in[3];
  float* out = (float*)d_out;

  // workspace: h (4096x2048 f32 = 32MB), then s1/s2/m/Z (4x16KB)
  float* h    = (float*)d_ws;
  float* s1   = h + (size_t)GN * GFOUT;
  float* s2   = s1 + GN;
  float* mrow = s2 + GN;
  float* zrow = mrow + GN;

  dim3 g1(GFOUT / 128, GN / 128);                   // 16 x 32
  gat_gemm_xw<<<g1, 256, 0, stream>>>(x, W, h);
  gat_scores<<<GN, 256, 0, stream>>>(h, a, s1, s2);
  gat_softmax_stats<<<GN, 256, 0, stream>>>(adj, s1, s2, mrow, zrow);
  dim3 g4(GFOUT / 64, GN / 128);                    // 32 x 32
  gat_aggregate<<<g4, 256, 0, stream>>>(adj, h, s1, s2, mrow, zrow, out);
}